// SkewSymMatrixExp_3650722201654
// MI455X (gfx1250) — compile-verified
//
#include <hip/hip_runtime.h>
#include <math.h>

// CDNA5 / gfx1250: wave32, WMMA f32 16x16x4.
typedef __attribute__((ext_vector_type(2))) float v2f;
typedef __attribute__((ext_vector_type(8))) float v8f;

#define D      64
#define LDA    65          // padded LDS row stride (floats): stride ≡ 1 mod 64 banks
#define NVEC   2016        // 64*63/2
#define NTHR   128         // 4 waves per workgroup

// dst(16x64 strip) = alpha * (A(64x64) * B(64x64)) + diag * I, one wave per strip.
// Four independent accumulator chains (one per 16-col tile) share each A fragment:
// 4-way WMMA ILP and A-operand LDS traffic cut 4x vs per-tile K-loops.
// All matrices in LDS with row stride LDA. EXEC must be all ones.
__device__ __forceinline__ void mm64(float* __restrict__ dst,
                                     const float* __restrict__ A,
                                     const float* __restrict__ B,
                                     float alpha, float diag,
                                     int wave, int lane) {
  const int half = lane >> 4;     // 0: lanes 0-15, 1: lanes 16-31
  const int lr   = lane & 15;
  const int m0   = wave * 16;

  v8f acc0 = {}, acc1 = {}, acc2 = {}, acc3 = {};
  // A 16x4 frag: lanes 0-15 rows M with {K0,K1}; lanes 16-31 rows M with {K2,K3}
  const float* arow = A + (m0 + lr) * LDA + 2 * half;
  // B 4x16 frag: row K striped across lanes (mirror of A layout)
  const float* brow = B + (2 * half) * LDA + lr;

  #pragma unroll
  for (int k = 0; k < 16; ++k) {
    v2f a, b0, b1, b2, b3;
    a.x = arow[4 * k];
    a.y = arow[4 * k + 1];
    const float* bp = brow + (4 * k) * LDA;
    b0.x = bp[0];       b0.y = bp[LDA];
    b1.x = bp[16];      b1.y = bp[LDA + 16];
    b2.x = bp[32];      b2.y = bp[LDA + 32];
    b3.x = bp[48];      b3.y = bp[LDA + 48];
    acc0 = __builtin_amdgcn_wmma_f32_16x16x4_f32(false, a, false, b0, (short)0, acc0, false, false);
    acc1 = __builtin_amdgcn_wmma_f32_16x16x4_f32(false, a, false, b1, (short)0, acc1, false, false);
    acc2 = __builtin_amdgcn_wmma_f32_16x16x4_f32(false, a, false, b2, (short)0, acc2, false, false);
    acc3 = __builtin_amdgcn_wmma_f32_16x16x4_f32(false, a, false, b3, (short)0, acc3, false, false);
  }

  // C/D layout: VGPR i -> rows m0+i (lanes 0-15) and m0+8+i (lanes 16-31).
  // Fused epilogue: dst = alpha*acc + diag*I.
  const int row = m0 + 8 * half;
  #pragma unroll
  for (int i = 0; i < 8; ++i) {
    const int r = row + i;
    float* drow = dst + r * LDA + lr;
    drow[0]  = acc0[i] * alpha + ((r == lr)      ? diag : 0.0f);
    drow[16] = acc1[i] * alpha + ((r == lr + 16) ? diag : 0.0f);
    drow[32] = acc2[i] * alpha + ((r == lr + 32) ? diag : 0.0f);
    drow[48] = acc3[i] * alpha + ((r == lr + 48) ? diag : 0.0f);
  }
}

__global__ void __launch_bounds__(NTHR)
skew_expm_kernel(const float* __restrict__ S_vec, float* __restrict__ out) {
  __shared__ float sA[D * LDA];   // raw skew matrix (unscaled)
  __shared__ float sP[D * LDA];   // ping
  __shared__ float sQ[D * LDA];   // pong
  __shared__ float sx[D], sy[D];  // power-iteration vectors
  __shared__ float s_fro2, s_n5, s_n6;

  const int tid  = threadIdx.x;
  const int wave = tid >> 5;
  const int lane = tid & 31;
  const long b   = blockIdx.x;
  const float* v = S_vec + b * (long)NVEC;

  if (tid == 0) { s_fro2 = 0.0f; s_n5 = 0.0f; s_n6 = 0.0f; }
  for (int i = tid; i < D * LDA; i += NTHR) sA[i] = 0.0f;
  __syncthreads();

  // Scatter strict-upper-triangle vector into skew-symmetric matrix:
  // S[i][j] = -v[p] (i<j), S[j][i] = +v[p]. Accumulate Frobenius norm^2.
  float local = 0.0f;
  for (int p = tid; p < NVEC; p += NTHR) {
    float fp = (float)p;
    int i = (int)((2.0f * D - 1.0f -
                   sqrtf((2.0f * D - 1.0f) * (2.0f * D - 1.0f) - 8.0f * fp)) * 0.5f);
    if (i < 0) i = 0;
    while (i > 0 && i * (2 * D - 1 - i) / 2 > p) --i;
    while ((i + 1) * (2 * D - 1 - (i + 1)) / 2 <= p) ++i;
    int j = i + 1 + (p - i * (2 * D - 1 - i) / 2);
    float val = v[p];
    sA[i * LDA + j] = -val;
    sA[j * LDA + i] =  val;
    local += val * val;
  }
  atomicAdd(&s_fro2, 2.0f * local);   // ds_add_f32
  __syncthreads();

  // --- Spectral norm estimate: 6 power-iteration matvecs (unnormalized). ---
  // For skew-symmetric A, ||A x|| <= lam_max ||x|| exactly, and the step-norm
  // ratio converges to lam_max from below; 1.3x safety covers non-convergence.
  {
    float* xv = sx;
    float* yv = sy;
    if (tid < D) xv[tid] = 1.0f;
    __syncthreads();
    #pragma unroll 1
    for (int it = 0; it < 6; ++it) {
      if (tid < D) {
        float s = 0.0f;
        const float* row = sA + tid * LDA;
        #pragma unroll 8
        for (int c = 0; c < D; ++c) s += row[c] * xv[c];
        yv[tid] = s;
      }
      __syncthreads();
      float* t = xv; xv = yv; yv = t;
    }
    // xv = x6, yv = x5
    if (tid < D) {
      atomicAdd(&s_n6, xv[tid] * xv[tid]);
      atomicAdd(&s_n5, yv[tid] * yv[tid]);
    }
    __syncthreads();
  }

  // Scaling: ||A/2^j||_2 <= bound/2^j <= 0.5. A itself stays unscaled;
  // the per-matmul alpha folds "scale" into the Horner coefficients.
  const float fro = sqrtf(s_fro2);
  float bound = fro;
  {
    const float est = sqrtf(s_n6 / fmaxf(s_n5, 1e-30f)) * 1.3f;
    if (est > 1e-20f && est < bound && isfinite(est)) bound = est;
  }
  int jsq = 0;
  while (ldexpf(bound, -jsq) > 0.5f && jsq < 24) ++jsq;
  const float scale = ldexpf(1.0f, -jsq);

  // Taylor degree 9 via Horner: T9 = I + (scale*A)/9.
  // Truncation error at ||scale*A|| <= 0.5: 0.5^10/10! ~ 2.7e-10 << fp32 eps.
  for (int i = tid; i < D * D; i += NTHR) {
    int r = i >> 6, c = i & 63;
    sP[r * LDA + c] = sA[r * LDA + c] * (scale * (1.0f / 9.0f)) + (r == c ? 1.0f : 0.0f);
  }
  __syncthreads();

  // T_k = I + (scale/k) * A * T_{k+1}, k = 8..1 (ping-pong P <-> Q)
  float* src = sP;
  float* dstb = sQ;
  #pragma unroll 1
  for (int k = 8; k >= 1; --k) {
    mm64(dstb, sA, src, scale / (float)k, 1.0f, wave, lane);
    __syncthreads();
    float* t = src; src = dstb; dstb = t;
  }

  // Squaring phase: P <- P*P, jsq times (ping-pong)
  #pragma unroll 1
  for (int s = 0; s < jsq; ++s) {
    mm64(dstb, src, src, 1.0f, 0.0f, wave, lane);
    __syncthreads();
    float* t = src; src = dstb; dstb = t;
  }

  // Write dense 64x64 result (coalesced)
  float* o = out + b * (long)(D * D);
  for (int i = tid; i < D * D; i += NTHR) {
    int r = i >> 6, c = i & 63;
    o[i] = src[r * LDA + c];
  }
}

extern "C" void kernel_launch(void* const* d_in, const int* in_sizes, int n_in,
                              void* d_out, int out_size, void* d_ws, size_t ws_size,
                              hipStream_t stream) {
  (void)n_in; (void)d_ws; (void)ws_size; (void)out_size;
  const float* S_vec = (const float*)d_in[0];
  float* out = (float*)d_out;
  const int batch = in_sizes[0] / NVEC;   // 4096
  skew_expm_kernel<<<batch, NTHR, 0, stream>>>(S_vec, out);
}